// EdgeAttentionAggregator_18923625906528
// MI455X (gfx1250) — compile-verified
//
#include <hip/hip_runtime.h>
#include <hip/hip_bf16.h>

#define N_NODES  50000
#define K_NB     32
#define IN_DIM   256
#define OUT_DIM  128
#define EDGE_DIM 64
#define LRELU_ALPHA 0.2f

typedef float v2f __attribute__((ext_vector_type(2)));
typedef float v8f __attribute__((ext_vector_type(8)));
typedef int   v4i_vs __attribute__((vector_size(16)));   // matches builtin param

// ---------------------------------------------------------------------------
// DS_SWIZZLE_B32 xor-butterfly: group-of-32 mode, offset = and_mask(0x1f) |
// or_mask(0)<<5 | xor_mask<<10.  Compile-time pattern, no index VALU setup.
// ---------------------------------------------------------------------------
template <int XOR_MASK>
__device__ __forceinline__ float swz_xor(float v) {
    return __int_as_float(
        __builtin_amdgcn_ds_swizzle(__float_as_int(v), 0x1f | (XOR_MASK << 10)));
}

__device__ __forceinline__ float wave_red_add16(float v) { // within 16-lane halves
    v += swz_xor<1>(v); v += swz_xor<2>(v); v += swz_xor<4>(v); v += swz_xor<8>(v);
    return v;
}
__device__ __forceinline__ float wave_red_add32(float v) {
    v += swz_xor<1>(v); v += swz_xor<2>(v); v += swz_xor<4>(v);
    v += swz_xor<8>(v); v += swz_xor<16>(v);
    return v;
}
__device__ __forceinline__ float wave_red_max32(float v) {
    v = fmaxf(v, swz_xor<1>(v));  v = fmaxf(v, swz_xor<2>(v));
    v = fmaxf(v, swz_xor<4>(v));  v = fmaxf(v, swz_xor<8>(v));
    v = fmaxf(v, swz_xor<16>(v));
    return v;
}

// ---------------------------------------------------------------------------
// GLOBAL_LOAD_ASYNC_TO_LDS_B128: direct memory->LDS DMA, tracked by ASYNCcnt.
// Builtin expects (int4 AS(1)* src, int4 AS(3)* dst, imm offset, imm cpol);
// cast through integers to materialize the address spaces (LDS offset = low
// 32 bits of the generic shared pointer, per the flat-aperture encoding).
// ---------------------------------------------------------------------------
__device__ __forceinline__ void async_copy_b128(const float* g, float* l) {
    auto gp = (__attribute__((address_space(1))) v4i_vs*)(uintptr_t)g;
    auto lp = (__attribute__((address_space(3))) v4i_vs*)(uint32_t)(uintptr_t)l;
    __builtin_amdgcn_global_load_async_to_lds_b128(gp, lp, 0, 0);
}

// ---------------------------------------------------------------------------
// Kernel 1: x = features @ W  via V_WMMA_F32_16X16X4_F32 (full fp32 precision)
// Fuses s_self = x . a[0:128] and s_nb = x . a[128:256] per row.
// Grid: N/16 blocks of 256 threads (8 waves). Wave w owns output cols [16w,16w+16).
// ---------------------------------------------------------------------------
__global__ __launch_bounds__(256)
void gat_gemm_kernel(const float* __restrict__ F, const float* __restrict__ W,
                     const float* __restrict__ a, float* __restrict__ x,
                     float* __restrict__ s_self, float* __restrict__ s_nb)
{
    const int m0   = blockIdx.x * 16;
    const int wave = threadIdx.x >> 5;
    const int lane = threadIdx.x & 31;
    const int lh   = lane & 15;
    const int hi   = lane >> 4;        // lane half: 0 or 1
    const int n0   = wave * 16;

    __shared__ float sacc[16];
    __shared__ float nacc[16];
    if (threadIdx.x < 16) { sacc[threadIdx.x] = 0.f; nacc[threadIdx.x] = 0.f; }
    __syncthreads();

    // A fragment (16x4 f32): VGPR0 = {K0 | K2}, VGPR1 = {K1 | K3} across lane halves
    const float* Arow = F + (size_t)(m0 + lh) * IN_DIM;

    v8f c = {};
    #pragma unroll 4
    for (int k0 = 0; k0 < IN_DIM; k0 += 4) {
        v2f av, bv;
        av.x = Arow[k0 + 2 * hi + 0];
        av.y = Arow[k0 + 2 * hi + 1];
        // B fragment (4x16 f32): row-striped across lanes within a VGPR
        bv.x = W[(size_t)(k0 + 2 * hi + 0) * OUT_DIM + n0 + lh];
        bv.y = W[(size_t)(k0 + 2 * hi + 1) * OUT_DIM + n0 + lh];
        c = __builtin_amdgcn_wmma_f32_16x16x4_f32(false, av, false, bv,
                                                  (short)0, c, false, false);
    }

    // C/D layout: VGPR r -> (M = r + 8*hi, N = n0 + lh)
    const float as = a[n0 + lh];             // a_self slice for this column
    const float an = a[OUT_DIM + n0 + lh];   // a_nb slice for this column
    #pragma unroll
    for (int r = 0; r < 8; ++r) {
        const int row = m0 + r + 8 * hi;
        const float v = c[r];
        x[(size_t)row * OUT_DIM + n0 + lh] = v;
        // reduce over the 16 lanes that share this output row
        const float ps = wave_red_add16(v * as);
        const float pn = wave_red_add16(v * an);
        if (lh == 0) {
            atomicAdd(&sacc[r + 8 * hi], ps);
            atomicAdd(&nacc[r + 8 * hi], pn);
        }
    }
    __syncthreads();
    if (threadIdx.x < 16) {
        s_self[m0 + threadIdx.x] = sacc[threadIdx.x];
        s_nb[m0 + threadIdx.x]   = nacc[threadIdx.x];
    }
}

// ---------------------------------------------------------------------------
// Kernel 2: per-node attention + aggregation. One 256-thread block per node.
// emb[n] (8KB) staged in LDS via async DMA; x gathers hit L2 (25.6 MB << 192 MB).
// ---------------------------------------------------------------------------
__global__ __launch_bounds__(256)
void gat_attn_kernel(const int* __restrict__ neigh, const float* __restrict__ emb,
                     const float* __restrict__ a, const float* __restrict__ x,
                     const float* __restrict__ s_self, const float* __restrict__ s_nb,
                     float* __restrict__ out)
{
    const int n    = blockIdx.x;
    const int t    = threadIdx.x;
    const int wave = t >> 5;
    const int lane = t & 31;

    __shared__ float semb[K_NB * EDGE_DIM];  // 8 KB: emb[n], reused twice
    __shared__ float att[K_NB];
    __shared__ int   snb[K_NB];
    __shared__ float sscore[K_NB];

    // async stage emb[n]: 2048 floats = 512 x b128, 2 per thread (ASYNCcnt)
    const float* src = emb + (size_t)n * (K_NB * EDGE_DIM);
    async_copy_b128(src + 4 * t,         semb + 4 * t);
    async_copy_b128(src + 4 * (t + 256), semb + 4 * (t + 256));

    // overlapped with the DMA: neighbor indices + gathered neighbor scores
    if (t < K_NB) {
        const int nk = neigh[(size_t)n * K_NB + t];
        snb[t]    = nk;
        sscore[t] = s_nb[nk];
    }
    __builtin_amdgcn_s_wait_asynccnt(0);
    __syncthreads();

    // edge scores: wave w handles k = 4w .. 4w+3 (32-lane dot over 64 dims)
    const float ae0 = a[2 * OUT_DIM + lane];
    const float ae1 = a[2 * OUT_DIM + 32 + lane];
    const float ss  = s_self[n];
    for (int kk = 0; kk < 4; ++kk) {
        const int k = wave * 4 + kk;
        float v = semb[k * EDGE_DIM + lane] * ae0
                + semb[k * EDGE_DIM + 32 + lane] * ae1;
        v = wave_red_add32(v);
        if (lane == 0) {
            const float e = ss + sscore[k] + v;
            att[k] = (e > 0.f) ? e : LRELU_ALPHA * e;   // leaky_relu
        }
    }
    __syncthreads();

    // softmax over the 32 scores, done by wave 0
    if (wave == 0) {
        const float e  = att[lane];
        const float m  = wave_red_max32(e);
        const float ex = __expf(e - m);
        const float sum = wave_red_add32(ex);
        att[lane] = ex / sum;
    }
    __syncthreads();

    float* orow = out + (size_t)n * (2 * OUT_DIM + EDGE_DIM);

    if (t < 128) {
        // h[d] = sum_k att[k] * x[neigh_k][d]   (gathers served from L2)
        float acc = 0.f;
        #pragma unroll 8
        for (int k = 0; k < K_NB; ++k)
            acc = fmaf(att[k], x[(size_t)snb[k] * OUT_DIM + t], acc);
        orow[OUT_DIM + t] = (acc > 0.f) ? acc : expm1f(acc);          // elu
    } else if (t < 192) {
        // h_e[e] = sum_k att[k] * emb[n][k][e]  (from LDS)
        const int ed = t - 128;
        float acc = 0.f;
        #pragma unroll 8
        for (int k = 0; k < K_NB; ++k)
            acc = fmaf(att[k], semb[k * EDGE_DIM + ed], acc);
        orow[2 * OUT_DIM + ed] = (acc > 0.f) ? acc : expm1f(acc);     // elu
    } else {
        // out[0:128] = elu(x[n]) : 64 threads x 2 columns
        const int d = t - 192;
        const float v0 = x[(size_t)n * OUT_DIM + d];
        const float v1 = x[(size_t)n * OUT_DIM + d + 64];
        orow[d]      = (v0 > 0.f) ? v0 : expm1f(v0);
        orow[d + 64] = (v1 > 0.f) ? v1 : expm1f(v1);
    }
}

// ---------------------------------------------------------------------------
extern "C" void kernel_launch(void* const* d_in, const int* in_sizes, int n_in,
                              void* d_out, int out_size, void* d_ws, size_t ws_size,
                              hipStream_t stream)
{
    const float* F     = (const float*)d_in[0];
    const int*   neigh = (const int*)d_in[1];
    const float* emb   = (const float*)d_in[2];
    const float* W     = (const float*)d_in[3];
    const float* a     = (const float*)d_in[4];
    float*       out   = (float*)d_out;

    // workspace: x (N*128 f32) | s_self (N) | s_nb (N)  ~ 26 MB
    float* x      = (float*)d_ws;
    float* s_self = x + (size_t)N_NODES * OUT_DIM;
    float* s_nb   = s_self + N_NODES;

    gat_gemm_kernel<<<N_NODES / 16, 256, 0, stream>>>(F, W, a, x, s_self, s_nb);
    gat_attn_kernel<<<N_NODES, 256, 0, stream>>>(neigh, emb, a, x, s_self, s_nb, out);
}